// Mahalanobis_mask_54743653155091
// MI455X (gfx1250) — compile-verified
//
#include <hip/hip_runtime.h>
#include <hip/hip_bf16.h>
#include <math.h>

// ---------------------------------------------------------------------------
// Problem constants (from reference): B=16, N=1024, D=512, FQ=257
// F padded to 272 = 17*16 so the WMMA K/N tiling is exact; padded columns are
// zero in the DFT weights and in w[], so Y's padded columns are exactly zero
// and the cross GEMM over K=272 is numerically identical to K=257.
// ---------------------------------------------------------------------------
#define BB   16
#define NN   1024
#define DD   512
#define FQ   257
#define FP   272              // padded frequency count (17 * 16)
#define ROWS (BB * NN)        // 16384 flattened (b, n) rows

typedef __attribute__((ext_vector_type(2))) float v2f;
typedef __attribute__((ext_vector_type(8))) float v8f;

#define TWO_PI 6.28318530717958647692f

// ---------------------------------------------------------------------------
// Kernel 0: w[f] = sum_k A[f, k]   (f < 257), zero for padded f
// ---------------------------------------------------------------------------
__global__ void wsum_kernel(const float* __restrict__ A, float* __restrict__ wv) {
    int f = blockIdx.x * blockDim.x + threadIdx.x;
    if (f >= FP) return;
    float s = 0.0f;
    if (f < FQ) {
        const float* row = A + (size_t)f * FQ;
        for (int k = 0; k < FQ; ++k) s += row[k];
    }
    wv[f] = s;
}

// ---------------------------------------------------------------------------
// Kernel 1: DFT weight tables Wc[d][f] = cos(2*pi*d*f/512),
//                             Ws[d][f] = sin(2*pi*d*f/512)   (sign irrelevant
// under the later abs()).  Padded f columns are zero.
// ---------------------------------------------------------------------------
__global__ void dft_table_kernel(float* __restrict__ Wc, float* __restrict__ Ws) {
    int idx = blockIdx.x * blockDim.x + threadIdx.x;
    if (idx >= DD * FP) return;
    int d = idx / FP;
    int f = idx % FP;
    if (f >= FQ) { Wc[idx] = 0.0f; Ws[idx] = 0.0f; return; }
    // reduce the phase argument mod 512 for accuracy
    int  r = (d * f) & (DD - 1);
    float t = TWO_PI * (float)r * (1.0f / (float)DD);
    Wc[idx] = __cosf(t);
    Ws[idx] = __sinf(t);
}

// ---------------------------------------------------------------------------
// Kernel 2: DFT GEMM + |.| + per-frequency scale.
//   Re = X * Wc,  Im = X * Ws   via V_WMMA_F32_16X16X4_F32 chains,
//   Y[r][f] = sqrt(Re^2 + Im^2) * w[f]
// One wave (32 threads) computes one 16x16 output tile, two accumulators
// (cos / sin) advanced together over K = 512 in steps of 4.
//
// WMMA f32 16x16x4 VGPR layouts (ISA 7.12.2):
//   A (16x4):  lane L -> m = L&15, h = L>>4; VGPR v holds A[m][2h + v]
//   B (4x16):  lane L -> n = L&15, h = L>>4; VGPR v holds B[2h + v][n]
//   C/D:       lane L -> n = L&15, h = L>>4; VGPR v holds D[v + 8h][n]
// ---------------------------------------------------------------------------
__global__ void dft_gemm_kernel(const float* __restrict__ X,
                                const float* __restrict__ Wc,
                                const float* __restrict__ Ws,
                                const float* __restrict__ wv,
                                float* __restrict__ Y) {
    const int lane = threadIdx.x;        // 0..31
    const int m    = lane & 15;
    const int h    = lane >> 4;
    const int row0 = blockIdx.x * 16;    // 0..16368
    const int col0 = blockIdx.y * 16;    // 0..256

    const float* xrow = X + (size_t)(row0 + m) * DD;

    v8f accC = {0.f, 0.f, 0.f, 0.f, 0.f, 0.f, 0.f, 0.f};
    v8f accS = {0.f, 0.f, 0.f, 0.f, 0.f, 0.f, 0.f, 0.f};

    for (int k = 0; k < DD; k += 4) {
        const int kk = k + 2 * h;
        v2f a;
        a.x = xrow[kk + 0];
        a.y = xrow[kk + 1];

        const float* wc = Wc + (size_t)kk * FP + col0 + m;
        v2f bc; bc.x = wc[0]; bc.y = wc[FP];

        const float* wsp = Ws + (size_t)kk * FP + col0 + m;
        v2f bs; bs.x = wsp[0]; bs.y = wsp[FP];

        accC = __builtin_amdgcn_wmma_f32_16x16x4_f32(false, a, false, bc,
                                                     (short)0, accC, false, false);
        accS = __builtin_amdgcn_wmma_f32_16x16x4_f32(false, a, false, bs,
                                                     (short)0, accS, false, false);
    }

    const float wf = wv[col0 + m];
#pragma unroll
    for (int v = 0; v < 8; ++v) {
        const int r  = row0 + v + 8 * h;
        const float re = accC[v];
        const float im = accS[v];
        Y[(size_t)r * FP + col0 + m] = sqrtf(re * re + im * im) * wf;
    }
}

// ---------------------------------------------------------------------------
// Kernel 3: sq[r] = sum_f Y[r][f]^2   (one wave per row, shfl reduce)
// ---------------------------------------------------------------------------
__global__ void sq_kernel(const float* __restrict__ Y, float* __restrict__ sq) {
    const int row  = blockIdx.x;
    const int lane = threadIdx.x;      // 32 threads
    const float* y = Y + (size_t)row * FP;
    float s = 0.0f;
    for (int f = lane; f < FP; f += 32) {
        float v = y[f];
        s += v * v;
    }
#pragma unroll
    for (int o = 16; o > 0; o >>= 1) s += __shfl_down(s, o, 32);
    if (lane == 0) sq[row] = s;
}

// ---------------------------------------------------------------------------
// Kernel 4: cross[b][i][j] = sum_f Y[b,i,f] * Y[b,j,f]  (batched Y * Y^T).
// One wave per 16x16 tile; B-matrix frag is a row of Y (K along memory),
// which matches the B layout with a contiguous float2 load per lane.
// ---------------------------------------------------------------------------
__global__ void cross_gemm_kernel(const float* __restrict__ Y,
                                  float* __restrict__ cross) {
    const int lane = threadIdx.x;
    const int m    = lane & 15;
    const int h    = lane >> 4;
    const int b    = blockIdx.z;
    const int i0   = blockIdx.x * 16;
    const int j0   = blockIdx.y * 16;

    const float* Yb   = Y + (size_t)b * NN * FP;
    const float* arow = Yb + (size_t)(i0 + m) * FP;   // A[m][k] = Y[i0+m][k]
    const float* brow = Yb + (size_t)(j0 + m) * FP;   // B[k][n] = Y[j0+n][k], n=m

    v8f acc = {0.f, 0.f, 0.f, 0.f, 0.f, 0.f, 0.f, 0.f};

    for (int k = 0; k < FP; k += 4) {
        const int kk = k + 2 * h;
        v2f a;  a.x  = arow[kk];  a.y  = arow[kk + 1];
        v2f bb; bb.x = brow[kk];  bb.y = brow[kk + 1];
        acc = __builtin_amdgcn_wmma_f32_16x16x4_f32(false, a, false, bb,
                                                    (short)0, acc, false, false);
    }

#pragma unroll
    for (int v = 0; v < 8; ++v) {
        const int i = i0 + v + 8 * h;
        const int j = j0 + m;
        cross[((size_t)b * NN + i) * NN + j] = acc[v];
    }
}

// ---------------------------------------------------------------------------
// Kernel 5: fused epilogue per (b, i) row:
//   dist[j]   = max(sq_i + sq_j - 2*cross, 0)
//   ed[j]     = (j==i) ? 0 : 1/(dist + 1e-10)
//   rmax      = max_j ed[j]
//   p[j]      = 0.99 * ed/(rmax+1e-10) + 0.01 * (j==i)
//   logit     = log(p/(1-p)); two gumbel draws (hash RNG); hard argmax.
// Straight-through forward value == hard mask.
// ---------------------------------------------------------------------------
__device__ __forceinline__ unsigned pcg_hash(unsigned x) {
    x = x * 747796405u + 2891336453u;
    unsigned w = ((x >> ((x >> 28u) + 4u)) ^ x) * 277803737u;
    return (w >> 22u) ^ w;
}

__device__ __forceinline__ float gumbel_from(unsigned seed) {
    unsigned hv = pcg_hash(seed);
    float u = ((float)hv + 0.5f) * 2.3283064365386963e-10f;  // (0,1)
    return -logf(-logf(u));
}

__global__ void mask_kernel(const float* __restrict__ cross,
                            const float* __restrict__ sq,
                            float* __restrict__ out) {
    const int row = blockIdx.x;           // b*N + i
    const int b   = row >> 10;
    const int i   = row & (NN - 1);
    const float* crow = cross + (size_t)row * NN;
    const float* sqb  = sq + ((size_t)b << 10);
    const float  sqi  = sq[row];

    __shared__ float smax[256];

    float m = 0.0f;
    for (int j = threadIdx.x; j < NN; j += 256) {
        float dist = fmaxf(sqi + sqb[j] - 2.0f * crow[j], 0.0f);
        float ed   = (j == i) ? 0.0f : 1.0f / (dist + 1e-10f);
        m = fmaxf(m, ed);
    }
    smax[threadIdx.x] = m;
    __syncthreads();
#pragma unroll
    for (int s = 128; s > 0; s >>= 1) {
        if (threadIdx.x < s)
            smax[threadIdx.x] = fmaxf(smax[threadIdx.x], smax[threadIdx.x + s]);
        __syncthreads();
    }
    const float rinv = 1.0f / (smax[0] + 1e-10f);

    float* orow = out + (size_t)row * NN;
    for (int j = threadIdx.x; j < NN; j += 256) {
        float dist = fmaxf(sqi + sqb[j] - 2.0f * crow[j], 0.0f);
        float ed   = (j == i) ? 0.0f : 1.0f / (dist + 1e-10f);
        float p    = 0.99f * ed * rinv + ((j == i) ? 0.01f : 0.0f);
        float logit = logf(p / (1.0f - p));
        unsigned base = (unsigned)(row) * 2048u + (unsigned)j * 2u;
        float g0 = gumbel_from(base + 0u);
        float g1 = gumbel_from(base + 1u);
        // class 0 wins  <=>  logit + g0 > -logit + g1
        orow[j] = (logit + g0 > -logit + g1) ? 1.0f : 0.0f;
    }
}

// ---------------------------------------------------------------------------
// Launcher
// ---------------------------------------------------------------------------
extern "C" void kernel_launch(void* const* d_in, const int* in_sizes, int n_in,
                              void* d_out, int out_size, void* d_ws, size_t ws_size,
                              hipStream_t stream) {
    const float* X = (const float*)d_in[0];   // [16, 1024, 512]
    const float* A = (const float*)d_in[1];   // [257, 257]
    float*       O = (float*)d_out;           // [16, 1, 1024, 1024]

    // Workspace carve-up (floats):
    float* ws    = (float*)d_ws;
    float* wv    = ws;                        // 272
    float* Wc    = wv + FP;                   // 512*272
    float* Wsin  = Wc + (size_t)DD * FP;      // 512*272
    float* Y     = Wsin + (size_t)DD * FP;    // 16384*272
    float* sq    = Y + (size_t)ROWS * FP;     // 16384
    float* cross = sq + ROWS;                 // 16*1024*1024

    // 0) w[f] = rowsum(A)
    wsum_kernel<<<(FP + 63) / 64, 64, 0, stream>>>(A, wv);

    // 1) DFT weight tables
    dft_table_kernel<<<(DD * FP + 255) / 256, 256, 0, stream>>>(Wc, Wsin);

    // 2) DFT GEMM -> |rfft| * w  (one wave / 16x16 tile; WMMA f32 16x16x4)
    {
        dim3 grid(ROWS / 16, FP / 16, 1);     // 1024 x 17
        dft_gemm_kernel<<<grid, 32, 0, stream>>>(X, Wc, Wsin, wv, Y);
    }

    // 3) sq[r] = ||Y_r||^2
    sq_kernel<<<ROWS, 32, 0, stream>>>(Y, sq);

    // 4) cross = Y * Y^T per batch (WMMA f32 16x16x4)
    {
        dim3 grid(NN / 16, NN / 16, BB);      // 64 x 64 x 16
        cross_gemm_kernel<<<grid, 32, 0, stream>>>(Y, cross);
    }

    // 5) fused rowmax + gumbel-argmax mask
    mask_kernel<<<ROWS, 256, 0, stream>>>(cross, sq, O);

    (void)in_sizes; (void)n_in; (void)out_size; (void)ws_size;
}